// ResidualQuantizer_17068200035053
// MI455X (gfx1250) — compile-verified
//
#include <hip/hip_runtime.h>

typedef __attribute__((ext_vector_type(2))) float v2f;
typedef __attribute__((ext_vector_type(8))) float v8f;

#define NUM_EMB 2048
#define DIM     64
#define HW      4096              // 64*64
#define NBATCH  8
#define NROWS   (NBATCH * HW)     // 32768 flattened z vectors
#define PAD     68                // LDS row stride (floats): float4-aligned, conflict-free
#define KTILE   32                // codes per iteration (two 16-wide WMMA chains)
#define NTILES  (NUM_EMB / KTILE) // 64

// ---------------------------------------------------------------------------
// Precompute ||w_k||^2 for all 2048 codebook rows into workspace.
// ---------------------------------------------------------------------------
__global__ void wnorm_kernel(const float* __restrict__ w, float* __restrict__ wn) {
    int k = blockIdx.x * blockDim.x + threadIdx.x;
    if (k < NUM_EMB) {
        const float4* row = (const float4*)(w + (size_t)k * DIM);
        float s = 0.0f;
#pragma unroll
        for (int i = 0; i < DIM / 4; ++i) {
            float4 v = row[i];
            s += v.x * v.x + v.y * v.y + v.z * v.z + v.w * v.w;
        }
        wn[k] = s;
    }
}

__global__ void zero_kernel(float* p) { p[0] = 0.0f; }

__global__ void finalize_kernel(const float* __restrict__ acc, float* __restrict__ out_loss) {
    // mean over NROWS*DIM elements, * COMMITMENT_COST(0.25)
    out_loss[0] = acc[0] * (0.25f / (float)(NROWS * DIM));
}

// ---------------------------------------------------------------------------
// Fused distance-GEMM + argmin + gather + loss.
// 256 threads = 8 waves; each wave owns 16 z-rows; loop 64 tiles of 32 codes.
// Double-buffered LDS staging, two independent WMMA accumulator chains.
// ---------------------------------------------------------------------------
__global__ __launch_bounds__(256) void vq_kernel(const float* __restrict__ z,
                                                 const float* __restrict__ w,
                                                 const float* __restrict__ wn,
                                                 float* __restrict__ out_idx,
                                                 float* __restrict__ out_q,
                                                 float* __restrict__ loss_acc) {
    __shared__ float lds_w[2][KTILE * PAD];   // 2 x 8.7 KB double buffer

    const int t    = threadIdx.x;
    const int wave = t >> 5;
    const int lane = t & 31;
    const int n0   = blockIdx.x * 128 + wave * 16;  // first z-row of this wave's tile
    const int b    = n0 >> 12;                      // n0 / 4096  (batch)
    const int hw0  = n0 & 4095;                     // position within 64x64 image
    const int m    = lane & 15;                     // A row / B col index for this lane
    const int hi   = lane >> 4;                     // K-half select (0: K0/K1, 1: K2/K3)

    // --- Load A tile: 16 rows x 64 chans of z, per WMMA f32 16x16x4 A layout ---
    // z layout (B,C,H,W): element (n, c) at z[(b*64 + c)*4096 + hw]
    v2f   a[16];
    float part = 0.0f;
    const float* zrow = z + (size_t)b * DIM * HW + hw0 + m;
#pragma unroll
    for (int ci = 0; ci < 16; ++ci) {
        int   c  = 4 * ci + 2 * hi;
        float x0 = zrow[(size_t)c * HW];
        float x1 = zrow[(size_t)(c + 1) * HW];
        a[ci].x = x0;
        a[ci].y = x1;
        part += x0 * x0 + x1 * x1;
    }
    // lanes m and m+16 each hold half the channels of row m
    float znorm = part + __shfl_xor(part, 16, 32);  // full ||z_m||^2, in lanes m, m+16

    float bv[8];
    int   bi[8];
#pragma unroll
    for (int r = 0; r < 8; ++r) { bv[r] = 3.4e38f; bi[r] = 0; }

    // --- staging assignment: thread t stages code row t/8, 8 channels (t%8)*8 ---
    const int    code_s = t >> 3;          // 0..31
    const int    c_s    = (t & 7) << 3;    // 0..56
    const float* wsrc   = w + (size_t)code_s * DIM + c_s;   // + kt*KTILE*DIM per tile
    float*       ldst0  = &lds_w[0][code_s * PAD + c_s];
    float*       ldst1  = &lds_w[1][code_s * PAD + c_s];

    {   // prologue: stage tile 0 into buffer 0
        float4 r0 = *(const float4*)(wsrc);
        float4 r1 = *(const float4*)(wsrc + 4);
        *(float4*)(ldst0)     = r0;
        *(float4*)(ldst0 + 4) = r1;
    }
    __syncthreads();

    for (int kt = 0; kt < NTILES; ++kt) {
        const int k0 = kt * KTILE;

        // --- register-prefetch next tile (overlaps with WMMA below) ---
        float4 n0v, n1v;
        const float* nsrc = wsrc + (size_t)(kt + 1) * (KTILE * DIM);
        if (kt < NTILES - 1) {
            n0v = *(const float4*)(nsrc);
            n1v = *(const float4*)(nsrc + 4);
        }
        if (kt < NTILES - 2) {
            // warm L2/L0 for the tile after that (global_prefetch_b8)
            __builtin_prefetch((const void*)(nsrc + (size_t)(KTILE * DIM)), 0, 1);
        }

        // --- two independent WMMA chains over the 32-code tile ---
        const float* lw = &lds_w[kt & 1][0];
        v8f acc0 = {};
        v8f acc1 = {};
#pragma unroll
        for (int ci = 0; ci < 16; ++ci) {
            int c = 4 * ci + 2 * hi;
            v2f b0, b1;
            b0.x = lw[m * PAD + c];
            b0.y = lw[m * PAD + c + 1];
            b1.x = lw[(m + 16) * PAD + c];
            b1.y = lw[(m + 16) * PAD + c + 1];
            acc0 = __builtin_amdgcn_wmma_f32_16x16x4_f32(
                false, a[ci], false, b0, (short)0, acc0, false, false);
            acc1 = __builtin_amdgcn_wmma_f32_16x16x4_f32(
                false, a[ci], false, b1, (short)0, acc1, false, false);
        }

        const float wn0   = wn[k0 + m];
        const float wn1   = wn[k0 + 16 + m];
        const int   code0 = k0 + m;
        const int   code1 = k0 + 16 + m;
#pragma unroll
        for (int r = 0; r < 8; ++r) {
            float s0 = wn0 - 2.0f * acc0[r];     // ||w||^2 - 2 z.w
            if (s0 < bv[r]) { bv[r] = s0; bi[r] = code0; }
            float s1 = wn1 - 2.0f * acc1[r];
            if (s1 < bv[r]) { bv[r] = s1; bi[r] = code1; }
        }

        // --- stash next tile into the other buffer, single barrier ---
        if (kt < NTILES - 1) {
            float* dst = (kt & 1) ? ldst0 : ldst1;
            *(float4*)(dst)     = n0v;
            *(float4*)(dst + 4) = n1v;
        }
        __syncthreads();
    }

    // --- cross-lane argmin within each 16-lane group (rows M = r + 8*hi) ---
#pragma unroll
    for (int r = 0; r < 8; ++r) {
        float v = bv[r];
        int   i = bi[r];
#pragma unroll
        for (int off = 1; off < 16; off <<= 1) {
            float ov = __shfl_xor(v, off, 32);
            int   oi = __shfl_xor(i, off, 32);
            if (ov < v || (ov == v && oi < i)) { v = ov; i = oi; }
        }
        bv[r] = v;
        bi[r] = i;
    }

    // --- broadcast per-row winners to all lanes ---
    int   rowIdx[16];
    float rowVal[16];
#pragma unroll
    for (int M = 0; M < 16; ++M) {
        int src   = (M < 8) ? 0 : 16;
        int r     = M & 7;
        rowIdx[M] = __shfl(bi[r], src, 32);
        rowVal[M] = __shfl(bv[r], src, 32);
    }

    // --- indices output (as float; exact for k < 2^24) ---
    if (lane < 16) out_idx[n0 + lane] = (float)rowIdx[lane];

    // --- gather quantized vectors; each lane writes 2 channels x 16 rows ---
    {
        const int c0 = lane * 2;
        float*    qb = out_q + (size_t)b * DIM * HW + hw0;
#pragma unroll
        for (int M = 0; M < 16; ++M) {
            const float2 qv = *(const float2*)(w + (size_t)rowIdx[M] * DIM + c0);
            qb[(size_t)c0 * HW + M]       = qv.x;
            qb[(size_t)(c0 + 1) * HW + M] = qv.y;
        }
    }

    // --- commitment loss: ||z_m - q_m||^2 == znorm_m + (||q||^2 - 2 z.q) ---
    float total = 0.0f;
#pragma unroll
    for (int M = 0; M < 16; ++M) {
        total += __shfl(znorm, M, 32);  // znorm for row M lives in lane M
        total += rowVal[M];
    }
    if (lane == 0) atomicAdd(loss_acc, total);
}

// ---------------------------------------------------------------------------
extern "C" void kernel_launch(void* const* d_in, const int* in_sizes, int n_in,
                              void* d_out, int out_size, void* d_ws, size_t ws_size,
                              hipStream_t stream) {
    (void)in_sizes; (void)n_in; (void)out_size; (void)ws_size;

    const float* z = (const float*)d_in[0];
    const float* w = (const float*)d_in[1];

    float* out      = (float*)d_out;
    float* out_idx  = out;                         // 32768
    float* out_q    = out + NROWS;                 // 2097152
    float* out_loss = out + NROWS + NROWS * DIM;   // 1 (last element)

    float* wn       = (float*)d_ws;                // 2048 floats
    float* loss_acc = wn + NUM_EMB;                // 1 float

    wnorm_kernel<<<NUM_EMB / 256, 256, 0, stream>>>(w, wn);
    zero_kernel<<<1, 1, 0, stream>>>(loss_acc);
    vq_kernel<<<NROWS / 128, 256, 0, stream>>>(z, w, wn, out_idx, out_q, loss_acc);
    finalize_kernel<<<1, 1, 0, stream>>>(loss_acc, out_loss);
}